// DescriptorDiversityLoss_42004780155257
// MI455X (gfx1250) — compile-verified
//
#include <hip/hip_runtime.h>
#include <hip/hip_bf16.h>

typedef float v2f __attribute__((ext_vector_type(2)));
typedef float v8f __attribute__((ext_vector_type(8)));

#define M_ROWS 8192
#define D_COLS 256
#define NBLK 128
#define ROWS_PER_BLK 64   // NBLK * ROWS_PER_BLK == M_ROWS

// Kernel 1: per-block partial column sums via V_WMMA_F32_16X16X4_F32
// (A = ones 16x4, B = 4x16 tile of X, C accumulates), plus partial sum of
// squares (trace term) via per-lane FMA + LDS tree reduction.
__global__ __launch_bounds__(512) void DescDiv_colsum_wmma(
    const float* __restrict__ X,
    float* __restrict__ colpart,   // [NBLK][D_COLS]
    float* __restrict__ sqpart) {  // [NBLK]
  const int tid    = threadIdx.x;
  const int wave   = tid >> 5;      // 0..15 -> column strip
  const int lane   = tid & 31;
  const int lane16 = lane & 15;
  const int hi     = lane >> 4;     // half-wave selects rows {0,1} vs {2,3}
  const int c0     = wave * 16;
  const int r0     = blockIdx.x * ROWS_PER_BLK;

  v2f a;
  a.x = 1.0f;                       // all-ones A => D[m,n] = sum_k B[k,n] + C
  a.y = 1.0f;
  v8f c = {};                       // f32 16x16 accumulator (8 VGPRs)

  float accsq = 0.0f;

  // lane (hi,lane16) covers B[k= 2*hi + {0,1}, n=lane16] of each 4x16 tile
  const float* base = X + (size_t)(r0 + hi * 2) * D_COLS + c0 + lane16;

#pragma unroll 4
  for (int t = 0; t < ROWS_PER_BLK / 4; ++t) {
    const float* p = base + (size_t)t * 4 * D_COLS;
    v2f b;
    b.x = p[0];                     // row  r0 + 4t + 2*hi
    b.y = p[D_COLS];                // row  r0 + 4t + 2*hi + 1
    accsq = fmaf(b.x, b.x, accsq);
    accsq = fmaf(b.y, b.y, accsq);
    // 8 args: (neg_a, A, neg_b, B, c_mod, C, reuse_a, reuse_b)
    c = __builtin_amdgcn_wmma_f32_16x16x4_f32(
        /*neg_a=*/false, a, /*neg_b=*/false, b,
        /*c_mod=*/(short)0, c, /*reuse_a=*/false, /*reuse_b=*/false);
  }

  // C/D layout: VGPR0, lanes 0..15 hold D[M=0, N=lane] = colsum(c0+lane),
  // replicated across all rows since A is all-ones. One half-wave writes.
  if (hi == 0) {
    colpart[(size_t)blockIdx.x * D_COLS + c0 + lane16] = c[0];
  }

  __shared__ float lsq[512];
  lsq[tid] = accsq;
  __syncthreads();
  for (int s = 256; s > 0; s >>= 1) {
    if (tid < s) lsq[tid] += lsq[tid + s];
    __syncthreads();
  }
  if (tid == 0) sqpart[blockIdx.x] = lsq[0];
}

// Kernel 2: s[d] = sum_g colpart[g][d];  out = |(||s||^2 - trace) / (M*(M-1))|
__global__ __launch_bounds__(256) void DescDiv_finalize(
    const float* __restrict__ colpart,
    const float* __restrict__ sqpart,
    float* __restrict__ out) {
  __shared__ float lds[256];
  const int t = threadIdx.x;

  float s = 0.0f;
  for (int g = 0; g < NBLK; ++g) s += colpart[(size_t)g * D_COLS + t];
  lds[t] = s * s;
  __syncthreads();
  for (int st = 128; st > 0; st >>= 1) {
    if (t < st) lds[t] += lds[t + st];
    __syncthreads();
  }
  const float ssum = lds[0];        // ||s||^2 = sum(sim)
  __syncthreads();

  lds[t] = (t < NBLK) ? sqpart[t] : 0.0f;
  __syncthreads();
  for (int st = 128; st > 0; st >>= 1) {
    if (t < st) lds[t] += lds[t + st];
    __syncthreads();
  }
  if (t == 0) {
    const float trace = lds[0];
    const float denom = (float)M_ROWS * (float)(M_ROWS - 1);  // 2^26 - 2^13, exact in f32
    out[0] = fabsf((ssum - trace) / denom);
  }
}

extern "C" void kernel_launch(void* const* d_in, const int* in_sizes, int n_in,
                              void* d_out, int out_size, void* d_ws, size_t ws_size,
                              hipStream_t stream) {
  (void)in_sizes; (void)n_in; (void)out_size; (void)ws_size;
  const float* X = (const float*)d_in[0];          // (16,512,256) f32, row-major
  float* colpart = (float*)d_ws;                   // NBLK*D_COLS f32 (128 KB)
  float* sqpart  = (float*)d_ws + (size_t)NBLK * D_COLS;  // NBLK f32

  DescDiv_colsum_wmma<<<NBLK, 512, 0, stream>>>(X, colpart, sqpart);
  DescDiv_finalize<<<1, 256, 0, stream>>>(colpart, sqpart, (float*)d_out);
}